// GraphConv_46145128628706
// MI455X (gfx1250) — compile-verified
//
#include <hip/hip_runtime.h>
#include <hip/hip_bf16.h>

typedef __attribute__((ext_vector_type(2))) float v2f;
typedef __attribute__((ext_vector_type(8))) float v8f;

#define D 128
#define LDS_STRIDE 132   // 132 mod 64 banks = 4 -> conflict-free B-fragment loads

// ---------------------------------------------------------------------------
// Kernel A: zero the accumulator (d_out), float4 stores.
// ---------------------------------------------------------------------------
__global__ void zero_f4(float4* __restrict__ out, int n4) {
    int i = blockIdx.x * blockDim.x + threadIdx.x;
    if (i < n4) out[i] = make_float4(0.f, 0.f, 0.f, 0.f);
}

// ---------------------------------------------------------------------------
// Kernel B: COO edge scatter.  One wave32 per edge; lane l handles the
// 4-float chunk [4l, 4l+4).  Gather of ego[col] is a single coalesced 512B
// read per wave; scatter is 4 global_atomic_add_f32 per lane, landing in L2
// (the 102 MB accumulator is L2-resident on MI455X's 192 MB L2).
// ---------------------------------------------------------------------------
__global__ void __launch_bounds__(256)
edge_scatter(const float* __restrict__ ego,
             const float* __restrict__ vals,
             const int*   __restrict__ rows,
             const int*   __restrict__ cols,
             float*       __restrict__ out,
             int n_edges) {
    int gid  = blockIdx.x * blockDim.x + threadIdx.x;
    int edge = gid >> 5;
    int lane = gid & 31;
    if (edge >= n_edges) return;

    int   col = cols[edge];           // same addr across wave -> 1 L0 request
    int   row = rows[edge];
    float v   = vals[edge];

    const float4* src = (const float4*)(ego + (size_t)col * D);
    float4 m = src[lane];

    float* dst = out + (size_t)row * D + lane * 4;
    atomicAdd(dst + 0, v * m.x);
    atomicAdd(dst + 1, v * m.y);
    atomicAdd(dst + 2, v * m.z);
    atomicAdd(dst + 3, v * m.w);
}

// ---------------------------------------------------------------------------
// Kernel C: in-place transform  out[tile] = out[tile] @ W^T  using
// V_WMMA_F32_16X16X4_F32.  W staged in LDS (padded stride).  Each wave owns a
// 16-row tile: 8 N-tiles of 16x16 f32 accumulators, K swept in steps of 4.
// In-place is safe: a wave reads all 128 columns of its 16 rows before any
// write, and waves own disjoint rows.
// ---------------------------------------------------------------------------
__global__ void __launch_bounds__(256)
wmma_transform(const float* __restrict__ W,
               float*       __restrict__ out,
               int n_tiles) {
    __shared__ float Wl[D * LDS_STRIDE];

    // Stage W (128x128 f32, row-major) into LDS with padded stride.
    for (int i = threadIdx.x; i < (D * D) / 4; i += 256) {
        int r = (i * 4) >> 7;
        int c = (i * 4) & (D - 1);
        float4 w4 = ((const float4*)W)[i];
        float* p = &Wl[r * LDS_STRIDE + c];
        p[0] = w4.x; p[1] = w4.y; p[2] = w4.z; p[3] = w4.w;
    }
    __syncthreads();

    int waveInBlock = threadIdx.x >> 5;
    int lane        = threadIdx.x & 31;
    int tile        = blockIdx.x * 8 + waveInBlock;
    if (tile >= n_tiles) return;           // wave-uniform: EXEC stays all-1s

    float* base = out + (size_t)tile * 16 * D;

    // ISA fragment layout (16x4 f32 A / 4x16 f32 B):
    //   lanes 0-15 : m = lane, K = {k0, k0+1}
    //   lanes 16-31: m = lane-16, K = {k0+2, k0+3}
    int m     = lane & 15;
    int kHalf = (lane >> 4) << 1;          // 0 or 2

    v8f acc[8] = {};

    for (int k0 = 0; k0 < D; k0 += 4) {
        v2f a = *(const v2f*)(base + m * D + k0 + kHalf);
#pragma unroll
        for (int nt = 0; nt < 8; nt++) {
            int n0 = nt * 16;
            v2f b = *(const v2f*)(&Wl[(n0 + m) * LDS_STRIDE + k0 + kHalf]);
            acc[nt] = __builtin_amdgcn_wmma_f32_16x16x4_f32(
                /*neg_a=*/false, a, /*neg_b=*/false, b,
                /*c_mod=*/(short)0, acc[nt],
                /*reuse_a=*/false, /*reuse_b=*/false);
        }
    }

    // C/D layout: VGPR i -> row i (lanes 0-15) / row i+8 (lanes 16-31), col = lane&15.
    int cm = (lane >> 4) << 3;             // 0 or 8
    int cn = lane & 15;
#pragma unroll
    for (int nt = 0; nt < 8; nt++) {
        int n0 = nt * 16;
#pragma unroll
        for (int i = 0; i < 8; i++) {
            base[(size_t)(cm + i) * D + n0 + cn] = acc[nt][i];
        }
    }
}

// ---------------------------------------------------------------------------
// Launcher.  Inputs (setup_inputs order):
//   0: ego_embeddings [N*128] f32   1: vals [E] f32   2: W [128*128] f32
//   3: rows [E] i32                 4: cols [E] i32   5: n_nodes (scalar)
// Output: [N,128] f32.
// ---------------------------------------------------------------------------
extern "C" void kernel_launch(void* const* d_in, const int* in_sizes, int n_in,
                              void* d_out, int out_size, void* d_ws, size_t ws_size,
                              hipStream_t stream) {
    const float* ego  = (const float*)d_in[0];
    const float* vals = (const float*)d_in[1];
    const float* W    = (const float*)d_in[2];
    const int*   rows = (const int*)d_in[3];
    const int*   cols = (const int*)d_in[4];
    float*       out  = (float*)d_out;

    int n_nodes = in_sizes[0] / D;     // 200000
    int n_edges = in_sizes[1];         // 6400000

    // A: zero the accumulator (d_out is poisoned by the harness).
    int n4 = n_nodes * (D / 4);
    zero_f4<<<(n4 + 255) / 256, 256, 0, stream>>>((float4*)out, n4);

    // B: edge gather/scale/scatter-add (one wave per edge).
    long long threads = (long long)n_edges * 32;
    int blocksB = (int)((threads + 255) / 256);
    edge_scatter<<<blocksB, 256, 0, stream>>>(ego, vals, rows, cols, out, n_edges);

    // C: in-place agg @ W^T with f32 WMMA (8 waves/block, 16 rows/wave).
    int n_tiles = n_nodes / 16;        // 12500
    wmma_transform<<<(n_tiles + 7) / 8, 256, 0, stream>>>(W, out, n_tiles);
}